// MedianPool3d_64682207477826
// MI455X (gfx1250) — compile-verified
//
#include <hip/hip_runtime.h>

// 3D median filter 3x3x3, pad=1, lower median (14th of 27), fp32.
// Shape: [1,1,128,256,256].
//
// Compute (per thread, along depth):
//   - keep sorted 3x3 layers in registers (sorted once, used by 3 outputs)
//   - one merge(9,9) shared by outputs d and d+1 (both contain layers {d,d+1})
//   - pruned odd-even-merge selection: element 13 of 27 in 25 min/max ops
// Data movement (gfx1250):
//   - branchless global_load_async_to_lds_b32 staging (ASYNCcnt); out-of-range
//     elements load from a zero word in d_ws (address select, no EXEC branches)
//   - 2 buffer pairs, software pipelined: ONE barrier per 2 output slices

#define TW 64               // tile width  (outputs per block in W)
#define TH 4                // tile height (outputs per block in H)
#define DC 32               // depths per block (multiple of 4)
#define SW (TW + 2)         // 66
#define SH (TH + 2)         // 6
#define SLICE (SW * SH)     // 396 floats per padded slice

__device__ __forceinline__ float mnf(float a, float b) { return fminf(a, b); }
__device__ __forceinline__ float mxf(float a, float b) { return fmaxf(a, b); }
__device__ __forceinline__ float md3(float a, float b, float c) {
    return __builtin_amdgcn_fmed3f(a, b, c);   // v_med3_f32
}

// ---- gfx1250 async global->LDS copy (one f32), tracked by ASYNCcnt ----
__device__ __forceinline__ void async_ld_b32(const float* g, float* l) {
    asm volatile("global_load_async_to_lds_b32 %0, %1, off"
                 :
                 : "v"((unsigned)(unsigned long long)(size_t)l),
                   "v"((unsigned long long)(size_t)g)
                 : "memory");
}
__device__ __forceinline__ void wait_async0() {
    asm volatile("s_wait_asynccnt 0x0" ::: "memory");
}

// sort3 via min3/med3/max3 (3 VALU ops)
__device__ __forceinline__ void sort3(float a, float b, float c,
                                      float& r0, float& r1, float& r2) {
    r0 = mnf(mnf(a, b), c);
    r1 = md3(a, b, c);
    r2 = mxf(mxf(a, b), c);
}

__device__ __forceinline__ void merge21(float x0, float x1, float y0,
                                        float& z0, float& z1, float& z2) {
    float t = mxf(x0, y0);
    z0 = mnf(x0, y0);
    z1 = mnf(t, x1);
    z2 = mxf(t, x1);
}

__device__ __forceinline__ void merge22(float x0, float x1, float y0, float y1,
                                        float& z0, float& z1, float& z2, float& z3) {
    float e = mxf(x0, y0);
    float o = mnf(x1, y1);
    z0 = mnf(x0, y0);
    z3 = mxf(x1, y1);
    z1 = mnf(e, o);
    z2 = mxf(e, o);
}

__device__ __forceinline__ void merge33(const float* x, const float* y, float* z) {
    float t = mxf(x[0], y[0]);
    float u = mnf(x[2], y[2]);
    float e0 = mnf(x[0], y[0]);
    float e1 = mnf(t, u);
    float e2 = mxf(t, u);
    float e3 = mxf(x[2], y[2]);
    float o0 = mnf(x[1], y[1]);
    float o1 = mxf(x[1], y[1]);
    z[0] = e0;
    z[1] = mnf(e1, o0); z[2] = mxf(e1, o0);
    z[3] = mnf(e2, o1); z[4] = mxf(e2, o1);
    z[5] = e3;
}

__device__ __forceinline__ void merge32(float x0, float x1, float x2,
                                        float y0, float y1, float* z) {
    float g0, g1, g2;
    merge21(x0, x2, y0, g0, g1, g2);
    float o0 = mnf(x1, y1), o1 = mxf(x1, y1);
    z[0] = g0;
    z[1] = mnf(g1, o0); z[2] = mxf(g1, o0);
    z[3] = mnf(g2, o1); z[4] = mxf(g2, o1);
}

__device__ __forceinline__ void merge31(float x0, float x1, float x2, float y0, float* z) {
    float f0, f1, f2;
    merge21(x0, x2, y0, f0, f1, f2);
    z[0] = f0;
    z[1] = mnf(f1, x1); z[2] = mxf(f1, x1);
    z[3] = f2;
}

// full sort of 9: 3x sort3 + merge(3,3) + merge(6,3)  (~45 ops)
__device__ __forceinline__ void sort9(const float* v, float* s) {
    float a[3], b[3], c[3];
    sort3(v[0], v[1], v[2], a[0], a[1], a[2]);
    sort3(v[3], v[4], v[5], b[0], b[1], b[2]);
    sort3(v[6], v[7], v[8], c[0], c[1], c[2]);
    float u[6];
    merge33(a, b, u);
    float E[5], O[4];
    merge32(u[0], u[2], u[4], c[0], c[2], E);
    merge31(u[1], u[3], u[5], c[1], O);
    s[0] = E[0];
    s[1] = mnf(E[1], O[0]); s[2] = mxf(E[1], O[0]);
    s[3] = mnf(E[2], O[1]); s[4] = mxf(E[2], O[1]);
    s[5] = mnf(E[3], O[2]); s[6] = mxf(E[3], O[2]);
    s[7] = mnf(E[4], O[3]); s[8] = mxf(E[4], O[3]);
}

__device__ __forceinline__ void merge44(const float* x, const float* y, float* z) {
    float E[4], O[4];
    merge22(x[0], x[2], y[0], y[2], E[0], E[1], E[2], E[3]);
    merge22(x[1], x[3], y[1], y[3], O[0], O[1], O[2], O[3]);
    z[0] = E[0];
    z[1] = mnf(E[1], O[0]); z[2] = mxf(E[1], O[0]);
    z[3] = mnf(E[2], O[1]); z[4] = mxf(E[2], O[1]);
    z[5] = mnf(E[3], O[2]); z[6] = mxf(E[3], O[2]);
    z[7] = O[3];
}

__device__ __forceinline__ void merge55(const float* x, const float* y, float* z) {
    float xe[3] = {x[0], x[2], x[4]}, ye[3] = {y[0], y[2], y[4]};
    float E[6];
    merge33(xe, ye, E);
    float O[4];
    merge22(x[1], x[3], y[1], y[3], O[0], O[1], O[2], O[3]);
    z[0] = E[0];
    z[1] = mnf(E[1], O[0]); z[2] = mxf(E[1], O[0]);
    z[3] = mnf(E[2], O[1]); z[4] = mxf(E[2], O[1]);
    z[5] = mnf(E[3], O[2]); z[6] = mxf(E[3], O[2]);
    z[7] = mnf(E[4], O[3]); z[8] = mxf(E[4], O[3]);
    z[9] = E[5];
}

// merge sorted a[9], b[9] -> s[18]  (30 CEs)
__device__ __forceinline__ void merge99(const float* a, const float* b, float* s) {
    float xe[5] = {a[0], a[2], a[4], a[6], a[8]};
    float ye[5] = {b[0], b[2], b[4], b[6], b[8]};
    float E[10];
    merge55(xe, ye, E);
    float xo[4] = {a[1], a[3], a[5], a[7]};
    float yo[4] = {b[1], b[3], b[5], b[7]};
    float O[8];
    merge44(xo, yo, O);
    s[0] = E[0];
#pragma unroll
    for (int i = 1; i <= 8; ++i) {
        s[2 * i - 1] = mnf(E[i], O[i - 1]);
        s[2 * i]     = mxf(E[i], O[i - 1]);
    }
    s[17] = E[9];
}

// element 13 (0-based) of merge(S[18], C[9]) — pruned odd-even merge, 25 ops
__device__ __forceinline__ float sel13(const float* S, const float* C) {
    float e3a  = mxf(mnf(mxf(S[0], C[0]), S[16]), mnf(S[8], C[8]));
    float o3a  = mnf(mxf(S[4], C[4]), S[12]);
    float e2a4 = mxf(e3a, o3a);
    float e3b  = mxf(mxf(S[2], C[2]), S[10]);
    float o3b  = mnf(mxf(S[6], C[6]), S[14]);
    float o2a3 = mnf(e3b, o3b);
    float E17  = mnf(e2a4, o2a3);
    float e3c  = mxf(mnf(mxf(S[1], C[1]), S[17]), S[9]);
    float o3c  = mnf(mxf(S[5], C[5]), S[13]);
    float e2b3 = mnf(e3c, o3c);
    float e3d  = mnf(mxf(S[3], C[3]), S[11]);
    float o3d  = mnf(S[7], C[7]);
    float o2b2 = mxf(e3d, o3d);
    float O16  = mxf(e2b3, o2b2);
    return mnf(E17, O16);
}

__global__ __launch_bounds__(256)
void median3d_kernel(const float* __restrict__ in, const float* __restrict__ zerow,
                     float* __restrict__ out, int D, int H, int W) {
    __shared__ float lds[4 * SLICE + 1];   // two slice-buffer pairs + dump word

    const int tid = threadIdx.x;
    const int w0 = blockIdx.x * TW;
    const int h0 = blockIdx.y * TH;
    const int d0 = blockIdx.z * DC;
    const int dEnd = d0 + DC;

    const size_t planeSz = (size_t)H * (size_t)W;

    // staging: thread t owns LDS slots t and t+256 of each 396-float slice
    const int r0 = tid / SW, c0 = tid % SW;
    const int t1 = tid + 256;
    const int r1 = t1 / SW, c1 = t1 % SW;
    const bool has1 = (t1 < SLICE);

    const int gh0 = h0 - 1 + r0, gw0 = w0 - 1 + c0;
    const int gh1 = h0 - 1 + r1, gw1 = w0 - 1 + c1;
    const bool ok0 = (gh0 >= 0) && (gh0 < H) && (gw0 >= 0) && (gw0 < W);
    const bool ok1 = has1 && (gh1 >= 0) && (gh1 < H) && (gw1 >= 0) && (gw1 < W);
    const size_t base0 = (size_t)(gh0 < 0 ? 0 : gh0) * W + (size_t)(gw0 < 0 ? 0 : gw0);
    const size_t base1 = (size_t)(gh1 < 0 ? 0 : gh1) * W + (size_t)(gw1 < 0 ? 0 : gw1);
    const int slot0 = r0 * SW + c0;
    const int slot1 = has1 ? (r1 * SW + c1) : (4 * SLICE);   // tail lanes -> dump word

    // branchless stage of input depth z into buffer `buf`:
    // every lane issues two async DMAs; out-of-range elements read the zero word
    auto stage = [&](int z, int buf) {
        const bool zin = (z >= 0) && (z < D);
        const float* gbase = in + (size_t)z * planeSz;
        const float* a0 = (zin && ok0) ? (gbase + base0) : zerow;
        const float* a1 = (zin && ok1) ? (gbase + base1) : zerow;
        float* lbase = &lds[0];
        async_ld_b32(a0, lbase + buf * SLICE + slot0);
        async_ld_b32(a1, (slot1 == 4 * SLICE) ? (lbase + slot1)
                                              : (lbase + buf * SLICE + slot1));
    };

    const int ty = tid >> 6;   // 0..3
    const int tx = tid & 63;   // 0..63
    const int oh = h0 + ty;
    const int ow = w0 + tx;

    auto taps = [&](int buf, float* v) {
#pragma unroll
        for (int j = 0; j < 3; ++j)
#pragma unroll
            for (int i = 0; i < 3; ++i)
                v[j * 3 + i] = lds[buf * SLICE + (ty + j) * SW + (tx + i)];
    };

    float sA[9], sB[9], sC[9], sD[9];

    // prologue: layers d0-1, d0 into bufs 2,3 ; body0 slices d0+1, d0+2 into bufs 0,1
    stage(d0 - 1, 2);
    stage(d0,     3);
    stage(d0 + 1, 0);
    stage(d0 + 2, 1);
    wait_async0();
    __syncthreads();
    {
        float v[9];
        taps(2, v); sort9(v, sA);
        taps(3, v); sort9(v, sB);
    }

    // body for outputs d, d+1. curPair: 0 -> bufs {0,1}, 1 -> bufs {2,3}.
    // A=sorted layer(d-1), B=layer(d); sorts C<-layer(d+1), Dn<-layer(d+2).
    auto body = [&](int d, int curPair, float* A, float* B, float* C, float* Dn) {
        wait_async0();          // drain staging issued by previous body
        __syncthreads();        // staged pair visible; old pair free to overwrite
        const int cb = curPair ? 2 : 0;
        const int nb = curPair ? 0 : 2;
        if (d + 2 < dEnd) {     // stage slices for the next body (overlaps compute)
            stage(d + 3, nb);
            stage(d + 4, nb + 1);
        }
        {
            float v[9];
            taps(cb,     v); sort9(v, C);
            taps(cb + 1, v); sort9(v, Dn);
        }
        float S[18];
        merge99(B, C, S);                       // shared by both outputs
        float m0 = sel13(S, A);                 // median at depth d
        float m1 = sel13(S, Dn);                // median at depth d+1
        __builtin_nontemporal_store(m0, &out[((size_t)d * H + oh) * W + ow]);
        __builtin_nontemporal_store(m1, &out[((size_t)(d + 1) * H + oh) * W + ow]);
    };

    for (int t = 0; t < DC; t += 4) {
        body(d0 + t,     0, sA, sB, sC, sD);
        body(d0 + t + 2, 1, sC, sD, sA, sB);
    }
}

extern "C" void kernel_launch(void* const* d_in, const int* in_sizes, int n_in,
                              void* d_out, int out_size, void* d_ws, size_t ws_size,
                              hipStream_t stream) {
    (void)in_sizes; (void)n_in; (void)out_size;
    const float* x = (const float*)d_in[0];
    float* out = (float*)d_out;

    // zero word for branchless out-of-range staging loads (graph-capturable)
    size_t zbytes = ws_size < 256 ? ws_size : 256;
    hipMemsetAsync(d_ws, 0, zbytes, stream);
    const float* zerow = (const float*)d_ws;

    const int D = 128, H = 256, W = 256;   // [1,1,128,256,256]
    dim3 grid(W / TW, H / TH, D / DC);     // (4, 64, 4)
    dim3 block(256);
    median3d_kernel<<<grid, block, 0, stream>>>(x, zerow, out, D, H, W);
}